// VQ2D_26938034881022
// MI455X (gfx1250) — compile-verified
//
#include <hip/hip_runtime.h>

// VQ 2-D nearest-code assignment, N=16.7M points, K=4 codes.
// Memory-bound streaming kernel: ~470 MB of traffic -> ~20us at 23.3 TB/s.
// WMMA is intentionally NOT used: a 16x16x4 WMMA would do 1024 MACs for 64
// useful dot products (6% utilization) plus cross-lane marshaling, all hidden
// behind the HBM wall anyway. The chip-rewarded path is b128 VMEM with
// gfx1250 non-temporal hints so the one-touch stream bypasses L2 retention.

typedef float v4f __attribute__((ext_vector_type(4)));
typedef float v2f __attribute__((ext_vector_type(2)));

__global__ void __launch_bounds__(256) vq2d_assign_kernel(
    const float* __restrict__ z,      // [N,2] flat
    const float* __restrict__ cb,     // [4,2] flat
    float* __restrict__ qgrad,        // [N,2] flat  (forward == z + (q - z))
    float* __restrict__ idxf,         // [N]   argmin as float 0..3
    float* __restrict__ q,            // [N,2] flat
    int nvec)                         // number of float4 chunks = N/2
{
    const int g = blockIdx.x * 256 + threadIdx.x;
    if (g >= nvec) return;

    // Codebook: 8 floats at a uniform address (hits L0/L2; 32B total).
    const float cx0 = cb[0], cy0 = cb[1];
    const float cx1 = cb[2], cy1 = cb[3];
    const float cx2 = cb[4], cy2 = cb[5];
    const float cx3 = cb[6], cy3 = cb[7];
    const float cx[4] = {cx0, cx1, cx2, cx3};
    const float cy[4] = {cy0, cy1, cy2, cy3};
    float cn[4];
#pragma unroll
    for (int k = 0; k < 4; ++k) cn[k] = cx[k] * cx[k] + cy[k] * cy[k];

    // One b128 NT load = two points for this lane; lane-contiguous -> fully
    // coalesced global_load_b128 th:NT.
    const v4f zz = __builtin_nontemporal_load((const v4f*)z + g);

    const float px[2] = {zz.x, zz.z};
    const float py[2] = {zz.y, zz.w};

    float qgx[2], qgy[2], qx[2], qy[2], bi[2];

#pragma unroll
    for (int p = 0; p < 2; ++p) {
        const float zx = px[p], zy = py[p];
        const float zn = zx * zx + zy * zy;

        // dists[k] = |z|^2 - 2 z.c_k + |c_k|^2, same algebra as the reference.
        // Strict '<' keeps first-min tie-breaking like jnp.argmin.
        float bestd = (zn - 2.0f * (zx * cx[0] + zy * cy[0])) + cn[0];
        float bx = cx[0], by = cy[0], bidx = 0.0f;
#pragma unroll
        for (int k = 1; k < 4; ++k) {
            const float d = (zn - 2.0f * (zx * cx[k] + zy * cy[k])) + cn[k];
            const bool lt = d < bestd;          // -> v_cmp + v_cndmask chain
            bestd = lt ? d : bestd;
            bx    = lt ? cx[k] : bx;
            by    = lt ? cy[k] : by;
            bidx  = lt ? (float)k : bidx;
        }
        qx[p] = bx;
        qy[p] = by;
        bi[p] = bidx;
        // Reproduce the reference's forward rounding: z + (q - z), not q.
        qgx[p] = zx + (bx - zx);
        qgy[p] = zy + (by - zy);
    }

    const v4f qgv = {qgx[0], qgy[0], qgx[1], qgy[1]};
    const v4f qv  = {qx[0],  qy[0],  qx[1],  qy[1]};
    const v2f iv  = {bi[0], bi[1]};

    // One-touch outputs: b128/b64 stores with NT temporal hint.
    __builtin_nontemporal_store(qgv, (v4f*)qgrad + g);
    __builtin_nontemporal_store(qv,  (v4f*)q + g);
    __builtin_nontemporal_store(iv,  (v2f*)idxf + g);
}

extern "C" void kernel_launch(void* const* d_in, const int* in_sizes, int n_in,
                              void* d_out, int out_size, void* d_ws, size_t ws_size,
                              hipStream_t stream) {
    const float* z  = (const float*)d_in[0];   // [N,2] float32
    const float* cb = (const float*)d_in[1];   // [4,2] float32

    const int zelems = in_sizes[0];            // N*2
    const int N      = zelems / 2;
    const int nvec   = zelems / 4;             // float4 chunks (2 points each)

    float* out   = (float*)d_out;              // 5N floats total
    float* qgrad = out;                        // [0, 2N)
    float* idxf  = out + (size_t)2 * N;        // [2N, 3N)
    float* q     = out + (size_t)3 * N;        // [3N, 5N)

    const int blocks = (nvec + 255) / 256;
    vq2d_assign_kernel<<<dim3(blocks), dim3(256), 0, stream>>>(
        z, cb, qgrad, idxf, q, nvec);
}